// CRF_52252572123524
// MI455X (gfx1250) — compile-verified
//
#include <hip/hip_runtime.h>
#include <math.h>

#define B_ 256
#define T_ 512
#define K_ 128
#define SA 132   // padded LDS row stride in floats (kills bank conflicts)

typedef float v2f __attribute__((ext_vector_type(2)));
typedef float v8f __attribute__((ext_vector_type(8)));

// ---------------------------------------------------------------------------
// Kernel 1: path score  (unary gather + transition gather), one block / batch
// ---------------------------------------------------------------------------
__global__ __launch_bounds__(128) void crf_score_kernel(
    const float* __restrict__ logits, const int* __restrict__ labels,
    const int* __restrict__ seq_lens, const float* __restrict__ trans,
    float* __restrict__ scores)
{
  const int b = blockIdx.x;
  const int L = seq_lens[b];
  float acc = 0.f;
  for (int t = threadIdx.x; t < T_; t += 128) {
    if (t < L) {
      const int lab = labels[b * T_ + t];
      acc += logits[((size_t)(b * T_ + t)) * K_ + lab];
      if (t >= 1) {
        const int labp = labels[b * T_ + t - 1];
        acc += trans[labp * K_ + lab];
      }
    }
  }
  __shared__ float red[128];
  red[threadIdx.x] = acc;
  __syncthreads();
  for (int s = 64; s > 0; s >>= 1) {
    if (threadIdx.x < s) red[threadIdx.x] += red[threadIdx.x + s];
    __syncthreads();
  }
  if (threadIdx.x == 0) scores[b] = red[0];
}

// ---------------------------------------------------------------------------
// Kernel 2: forward algorithm as exp-domain GEMM on V_WMMA_F32_16X16X4_F32.
// Block = 16 batch rows, 256 threads = 8 waves; wave w owns state cols
// [16w,16w+16). exp(trans) B-fragments are loop-invariant -> registers.
// ---------------------------------------------------------------------------
__global__ __launch_bounds__(256) void crf_forward_kernel(
    const float* __restrict__ logits, const int* __restrict__ seq_lens,
    const float* __restrict__ trans, const float* __restrict__ scores,
    float* __restrict__ out)
{
  __shared__ float s_alpha[16 * SA];
  __shared__ float s_aexp [16 * SA];
  __shared__ float s_m[16];
  __shared__ int   s_len[16];

  const int tid   = threadIdx.x;
  const int wave  = tid >> 5;
  const int lane  = tid & 31;
  const int bbase = blockIdx.x * 16;

  // ---- loop-invariant B fragments: exp(trans) tile for this wave's columns.
  // B (4x16 f32) layout mirrors A: lanes 0-15 hold K rows k,k+1; lanes 16-31
  // hold K rows k+2,k+3 (one row per VGPR half).
  const int colB = (wave << 4) + (lane & 15);
  const int kk   = (lane >> 4) << 1;     // 0 or 2
  v2f Bfrag[32];
#pragma unroll
  for (int kt = 0; kt < 32; ++kt) {
    const int kr = (kt << 2) + kk;
    v2f b;
    b.x = expf(trans[kr * K_ + colB]);
    b.y = expf(trans[(kr + 1) * K_ + colB]);
    Bfrag[kt] = b;
  }

  // ---- reduction mapping: 16 threads per batch row
  const int rrow = tid >> 4;   // 0..15
  const int j0   = tid & 15;

  if (tid < 16) s_len[tid] = seq_lens[bbase + tid];

  // alpha0 = logits[:, 0, :]
#pragma unroll
  for (int k = 0; k < 8; ++k) {
    const int j = j0 + 16 * k;
    s_alpha[rrow * SA + j] = logits[((size_t)(bbase + rrow) * T_) * K_ + j];
  }
  __syncthreads();

  // C/D tile mapping (16x16 f32): lane half selects M base 0 or 8
  const int crowbase = (lane >> 4) << 3;   // 0 or 8
  const int ccol     = (wave << 4) + (lane & 15);
  const int arow     = lane & 15;

  for (int t = 1; t < T_; ++t) {
    // (1) per-row max via 16-lane butterfly (all 16 lanes end with the max)
    float pm = s_alpha[rrow * SA + j0];
#pragma unroll
    for (int k = 1; k < 8; ++k)
      pm = fmaxf(pm, s_alpha[rrow * SA + j0 + 16 * k]);
#pragma unroll
    for (int d = 1; d < 16; d <<= 1)
      pm = fmaxf(pm, __shfl_xor(pm, d, 32));
    if (j0 == 0) s_m[rrow] = pm;

    // (2) normalized exponentials
#pragma unroll
    for (int k = 0; k < 8; ++k) {
      const int j = j0 + 16 * k;
      s_aexp[rrow * SA + j] = expf(s_alpha[rrow * SA + j] - pm);
    }
    __syncthreads();   // aexp ready; all alpha reads of this step done

    // prefetch next-step logits while the GEMM runs
    if (t + 1 < T_) {
#pragma unroll
      for (int v = 0; v < 8; v += 4) {
        const int row = crowbase + v;
        __builtin_prefetch(
            &logits[((size_t)(bbase + row) * T_ + (t + 1)) * K_ + ccol], 0, 1);
      }
    }

    // (3) S = Aexp * exp(trans) : 32 chained wmma_f32_16x16x4, 4 accumulators
    v8f acc0 = {}, acc1 = {}, acc2 = {}, acc3 = {};
#pragma unroll
    for (int kt = 0; kt < 32; kt += 4) {
      v2f a0 = *(const v2f*)&s_aexp[arow * SA + (kt + 0) * 4 + kk];
      v2f a1 = *(const v2f*)&s_aexp[arow * SA + (kt + 1) * 4 + kk];
      v2f a2 = *(const v2f*)&s_aexp[arow * SA + (kt + 2) * 4 + kk];
      v2f a3 = *(const v2f*)&s_aexp[arow * SA + (kt + 3) * 4 + kk];
      acc0 = __builtin_amdgcn_wmma_f32_16x16x4_f32(false, a0, false, Bfrag[kt + 0],
                                                   (short)0, acc0, false, false);
      acc1 = __builtin_amdgcn_wmma_f32_16x16x4_f32(false, a1, false, Bfrag[kt + 1],
                                                   (short)0, acc1, false, false);
      acc2 = __builtin_amdgcn_wmma_f32_16x16x4_f32(false, a2, false, Bfrag[kt + 2],
                                                   (short)0, acc2, false, false);
      acc3 = __builtin_amdgcn_wmma_f32_16x16x4_f32(false, a3, false, Bfrag[kt + 3],
                                                   (short)0, acc3, false, false);
    }
    v8f s = (acc0 + acc1) + (acc2 + acc3);

    // (4) alpha' = m + log(S) + logit[t], masked by seq_len (keep old if past)
#pragma unroll
    for (int v = 0; v < 8; ++v) {
      const int row = crowbase + v;
      if (t < s_len[row]) {
        const float lg =
            logits[((size_t)(bbase + row) * T_ + t) * K_ + ccol];
        s_alpha[row * SA + ccol] = s_m[row] + logf(s[v]) + lg;
      }
    }
    __syncthreads();   // alpha updated before next step's reads
  }

  // ---- log Z per batch row, then NLL contribution
  float pm = s_alpha[rrow * SA + j0];
#pragma unroll
  for (int k = 1; k < 8; ++k)
    pm = fmaxf(pm, s_alpha[rrow * SA + j0 + 16 * k]);
#pragma unroll
  for (int d = 1; d < 16; d <<= 1)
    pm = fmaxf(pm, __shfl_xor(pm, d, 32));

  float ps = 0.f;
#pragma unroll
  for (int k = 0; k < 8; ++k)
    ps += expf(s_alpha[rrow * SA + j0 + 16 * k] - pm);
#pragma unroll
  for (int d = 1; d < 16; d <<= 1)
    ps += __shfl_xor(ps, d, 32);

  if (j0 == 0) {
    const float log_z = pm + logf(ps);
    atomicAdd(out, log_z - scores[bbase + rrow]);
  }
}

// ---------------------------------------------------------------------------
extern "C" void kernel_launch(void* const* d_in, const int* in_sizes, int n_in,
                              void* d_out, int out_size, void* d_ws, size_t ws_size,
                              hipStream_t stream) {
  const float* logits   = (const float*)d_in[0];
  const int*   labels   = (const int*)d_in[1];
  const int*   seq_lens = (const int*)d_in[2];
  const float* trans    = (const float*)d_in[3];
  float* out    = (float*)d_out;
  float* scores = (float*)d_ws;   // 256 floats of scratch

  hipMemsetAsync(out, 0, sizeof(float), stream);
  crf_score_kernel<<<B_, 128, 0, stream>>>(logits, labels, seq_lens, trans,
                                           scores);
  crf_forward_kernel<<<B_ / 16, 256, 0, stream>>>(logits, seq_lens, trans,
                                                  scores, out);
}